// CenterLoss_52570399703413
// MI455X (gfx1250) — compile-verified
//
#include <hip/hip_runtime.h>

// CenterLoss on MI455X (gfx1250, wave32).
// Traffic ~1 MB total -> pure latency-bound. Strategy:
//   Kernel 1: one wave32 per batch row; coalesced float4 (b128) loads of
//             x[row] and centers[label[row]], fmaf accumulate of diff^2,
//             5-step shfl_xor wave reduction, clamp, write dist[row] to d_ws.
//   Kernel 2: single wave32 folds the 256 per-row dists with 4 chained
//             V_WMMA_F32_16X16X4_F32 ops (A = all-ones), then a 4-step
//             shuffle + scale by 1/BATCH writes the scalar mean.
// Deterministic (no float atomics), d_out written unconditionally.

#define BATCH     256
#define FEAT      512
#define CLAMP_MIN 1e-12f
#define CLAMP_MAX 1e12f

typedef __attribute__((ext_vector_type(2))) float v2f;
typedef __attribute__((ext_vector_type(8))) float v8f;

__global__ __launch_bounds__(256) void center_dist_kernel(
    const float* __restrict__ x,
    const int*   __restrict__ label,
    const float* __restrict__ centers,
    float*       __restrict__ dist) {
  const int wave = (int)(blockIdx.x * blockDim.x + threadIdx.x) >> 5;
  const int lane = (int)threadIdx.x & 31;
  if (wave >= BATCH) return;
  const int row = wave;

  const long long crow = (long long)label[row] * (long long)FEAT;
  const float4* __restrict__ xr = (const float4*)(x + (long long)row * FEAT);
  const float4* __restrict__ cr = (const float4*)(centers + crow);

  float acc = 0.0f;
#pragma unroll
  for (int i = 0; i < 4; ++i) {           // 4 x (32 lanes x float4) = 512 floats
    const int idx = i * 32 + lane;        // consecutive lanes -> contiguous b128
    float4 a = xr[idx];
    float4 b = cr[idx];
    float dx = a.x - b.x, dy = a.y - b.y, dz = a.z - b.z, dw = a.w - b.w;
    acc = fmaf(dx, dx, acc);
    acc = fmaf(dy, dy, acc);
    acc = fmaf(dz, dz, acc);
    acc = fmaf(dw, dw, acc);
  }
#pragma unroll
  for (int off = 16; off >= 1; off >>= 1)
    acc += __shfl_xor(acc, off, 32);

  if (lane == 0) {
    acc = fminf(fmaxf(acc, CLAMP_MIN), CLAMP_MAX);   // per-row clamp (pre-mean)
    dist[row] = acc;
  }
}

// Single-wave mean of 256 f32 via V_WMMA_F32_16X16X4_F32.
// A = ones(16x4) => D[i][j] = sum_k B[k][j] + C[i][j]; chaining 4 B-tiles that
// jointly cover all 256 inputs makes every column sum a disjoint partial sum,
// so sum_j D[0][j] == sum of all 256 inputs (exact wrt layout).
__global__ __launch_bounds__(32) void mean_wmma_kernel(
    const float* __restrict__ dist,
    float*       __restrict__ out) {
  const int lane = (int)threadIdx.x;      // 0..31, no divergence before WMMA

  float w[8];
#pragma unroll
  for (int k = 0; k < 8; ++k)
    w[k] = dist[lane + 32 * k];

  v2f ones = {1.0f, 1.0f};                // every element of A == 1.0
  v8f c = {};
#pragma unroll
  for (int t = 0; t < 4; ++t) {
    v2f b = {w[2 * t], w[2 * t + 1]};     // one 4x16 f32 B operand (2 VGPRs)
    // (neg_a, A, neg_b, B, c_mod, C, reuse_a, reuse_b)
    c = __builtin_amdgcn_wmma_f32_16x16x4_f32(
        false, ones, false, b, (short)0, c, false, false);
  }

  float s = c[0];                         // lane j (j<16): column sum j
#pragma unroll
  for (int off = 8; off >= 1; off >>= 1)  // fold lanes 0..15 -> lane 0
    s += __shfl_xor(s, off, 32);

  if (lane == 0)
    out[0] = s * (1.0f / (float)BATCH);
}

extern "C" void kernel_launch(void* const* d_in, const int* in_sizes, int n_in,
                              void* d_out, int out_size, void* d_ws, size_t ws_size,
                              hipStream_t stream) {
  const float* x       = (const float*)d_in[0];
  const int*   label   = (const int*)  d_in[1];
  const float* centers = (const float*)d_in[2];
  float*       dist    = (float*)d_ws;        // 256 floats of scratch
  float*       outp    = (float*)d_out;

  // 32 blocks x 8 waves = 256 waves, one per batch row.
  center_dist_kernel<<<32, 256, 0, stream>>>(x, label, centers, dist);
  mean_wmma_kernel<<<1, 32, 0, stream>>>(dist, outp);
}